// Model_34119220199995
// MI455X (gfx1250) — compile-verified
//
#include <hip/hip_runtime.h>
#include <hip/hip_bf16.h>

typedef __attribute__((ext_vector_type(2))) float v2f;
typedef __attribute__((ext_vector_type(4))) float v4f;
typedef __attribute__((ext_vector_type(8))) float v8f;
typedef unsigned long long u64;
typedef unsigned int u32;

#define WMMA4(a, b, c) \
  __builtin_amdgcn_wmma_f32_16x16x4_f32(false, (a), false, (b), (short)0, (c), false, false)

#define BATCH 2048
#define NREF  200000
#define NSPLIT 8
#define SPLITN (NREF / NSPLIT)   // 25000
#define TILE 128                 // 8 waves x 16 cols
#define TOPK 16

#define KEY_INF 0xFFFFFFFFFFFFFFFFull

// monotone float <-> sortable u32
__device__ __forceinline__ u32 fkey(float d) {
  u32 b = __float_as_uint(d);
  return (b & 0x80000000u) ? ~b : (b | 0x80000000u);
}
__device__ __forceinline__ float unfkey(u32 k) {
  return __uint_as_float((k & 0x80000000u) ? (k ^ 0x80000000u) : ~k);
}

// ---------------- Stage 1: q = sigmoid(relu(x @ Wc[:,:,3]^T + bc) @ Wl^T + bl)
__global__ __launch_bounds__(256) void qk(const float* __restrict__ x,
                                          const float* __restrict__ Wc,
                                          const float* __restrict__ bc,
                                          const float* __restrict__ Wl,
                                          const float* __restrict__ bl,
                                          float* __restrict__ q,
                                          float* __restrict__ qq) {
  int b = blockIdx.x * blockDim.x + threadIdx.x;
  if (b >= BATCH) return;
  float xr[25];
  #pragma unroll
  for (int i = 0; i < 25; ++i) xr[i] = x[b * 25 + i];
  float h[16];
  #pragma unroll
  for (int o = 0; o < 16; ++o) {
    float s = bc[o];
    #pragma unroll
    for (int i = 0; i < 25; ++i) s += xr[i] * Wc[o * 200 + i * 8 + 3];
    h[o] = s > 0.f ? s : 0.f;
  }
  float ss = 0.f;
  #pragma unroll
  for (int d = 0; d < 8; ++d) {
    float z = bl[d];
    #pragma unroll
    for (int o = 0; o < 16; ++o) z += h[o] * Wl[d * 16 + o];
    float s = 1.f / (1.f + __expf(-z));
    q[b * 8 + d] = s;
    ss += s * s;
  }
  qq[b] = ss;
}

// ---------------- Stage 2: WMMA distance tiles + per-split top-16 (packed keys)
// grid: (NSPLIT, BATCH/16); block: 256 (8 waves). Each wave makes a 16x16
// d2 tile per iteration with 3x v_wmma_f32_16x16x4_f32 (augmented K=10 trick).
__global__ __launch_bounds__(256) void distk(const float* __restrict__ q,
                                             const float* __restrict__ qq,
                                             const float* __restrict__ ref,
                                             u64* __restrict__ outk) {
  const int g    = blockIdx.y;          // row group (16 batch rows)
  const int s    = blockIdx.x;          // N split
  const int t    = threadIdx.x;
  const int lane = t & 31;
  const int wv   = t >> 5;              // wave 0..7
  const int m    = lane & 15;           // tile row (A) / tile col (B)
  const int hi   = lane >> 4;           // half-wave: K offset selector
  const int base = hi * 2;

  __shared__ float S[16 * 132];         // d2 tile, row-major, padded
  __shared__ u64   MK[16 * 16 * TOPK];  // merge: per row, 16 sorted key lists

  // --- A matrix registers (constant for whole block): q' = (-2q, qq, 1)
  const int row = g * 16 + m;
  const float* qr = q + row * 8;
  float q0 = qr[base], q1 = qr[base + 1], q4 = qr[base + 4], q5 = qr[base + 5];
  v2f a0 = { -2.f * q0, -2.f * q1 };    // K = base, base+1
  v2f a4 = { -2.f * q4, -2.f * q5 };    // K = base+4, base+5
  float qv = qq[row];
  v2f a8 = { hi ? 0.f : qv, hi ? 0.f : 1.f };  // K = 8,9 (hi=0) / 10,11 (=0)

  const int nStart = s * SPLITN;
  const int nEnd   = nStart + SPLITN;
  const int nTiles = (SPLITN + TILE - 1) / TILE;

  // per-thread top-16 as packed keys (sorted ascending, registers)
  u64 bd[TOPK];
  #pragma unroll
  for (int k = 0; k < TOPK; ++k) bd[k] = KEY_INF;
  const int r  = t >> 4;   // selection row 0..15
  const int c0 = t & 15;   // selection col block (8 contiguous cols)

  for (int tile = 0; tile < nTiles; ++tile) {
    // --- build B column r' = (r, 1, rr) for col n
    int n  = nStart + tile * TILE + wv * 16 + m;
    int nc = n < NREF ? n : NREF - 1;
    const float* rp = ref + nc * 8 + base;
    v2f p0 = { rp[0], rp[1] };          // K = base, base+1
    v2f p1 = { rp[4], rp[5] };          // K = base+4, base+5
    float pp = p0.x * p0.x + p0.y * p0.y + p1.x * p1.x + p1.y * p1.y;
    pp += __shfl_xor(pp, 16, 32);       // rr = ||r_n||^2 (both halves)
    v2f b8 = { hi ? 0.f : 1.f, hi ? 0.f : pp };

    v8f acc = {0.f, 0.f, 0.f, 0.f, 0.f, 0.f, 0.f, 0.f};
    acc = WMMA4(a0, p0, acc);
    acc = WMMA4(a4, p1, acc);
    acc = WMMA4(a8, b8, acc);           // acc = d2[m'][c] tile

    // D layout: lane l, vgpr v -> row v + 8*(l>=16), col l&15.
    // OOB columns poisoned with +INF at write time (no branch in reader).
    bool valid = n < nEnd;
    int col = wv * 16 + m;
    #pragma unroll
    for (int v = 0; v < 8; ++v)
      S[(v + 8 * hi) * 132 + col] = valid ? acc[v] : 3.0e38f;
    __syncthreads();

    // --- selection: 16 threads per row, 8 contiguous cols each, 2x b128 reads
    int nb = nStart + tile * TILE + c0 * 8;
    v4f x0 = *(const v4f*)&S[r * 132 + c0 * 8];
    v4f x1 = *(const v4f*)&S[r * 132 + c0 * 8 + 4];
    #pragma unroll
    for (int j = 0; j < 8; ++j) {
      float val = (j < 4) ? x0[j] : x1[j - 4];
      u64 key = ((u64)fkey(val) << 32) | (u32)(nb + j);
      if (key < bd[TOPK - 1]) {         // threshold: mostly rejects
        u64 c = key;
        #pragma unroll
        for (int p = 0; p < TOPK; ++p) {
          u64 old = bd[p];
          bool sw = c < old;            // branchless: min/max ladder
          bd[p] = sw ? c : old;
          c     = sw ? old : c;
        }
      }
    }
    __syncthreads();
  }

  // --- per-block merge: 16 sorted lists per row -> top-16 per (row, split)
  #pragma unroll
  for (int k = 0; k < TOPK; ++k)
    MK[(r * 16 + c0) * TOPK + k] = bd[k];
  __syncthreads();
  if (c0 == 0) {
    int head[16];
    for (int i = 0; i < 16; ++i) head[i] = 0;
    u64* ok = outk + ((g * 16 + r) * NSPLIT + s) * TOPK;
    for (int k = 0; k < TOPK; ++k) {
      u64 bv = KEY_INF; int bsel = 0;
      #pragma unroll
      for (int l = 0; l < 16; ++l) {
        u64 v = MK[(r * 16 + l) * TOPK + head[l]];
        if (v < bv) { bv = v; bsel = l; }
      }
      ok[k] = bv;
      head[bsel]++;
    }
  }
}

// ---------------- Stage 3: per row, 8-way merge of sorted 16-lists
__global__ __launch_bounds__(256) void mergek(const u64* __restrict__ keys,
                                              const int* __restrict__ ref_y,
                                              float* __restrict__ out) {
  int row = blockIdx.x * blockDim.x + threadIdx.x;
  if (row >= BATCH) return;
  const u64* kr = keys + row * (NSPLIT * TOPK);
  int head[NSPLIT];
  #pragma unroll
  for (int i = 0; i < NSPLIT; ++i) head[i] = 0;
  float* outd = out + row * TOPK;
  float* outl = out + BATCH * TOPK + row * TOPK;
  for (int k = 0; k < TOPK; ++k) {
    u64 bv = KEY_INF; int bsel = 0;
    #pragma unroll
    for (int l = 0; l < NSPLIT; ++l) {
      u64 v = kr[l * TOPK + head[l]];
      if (v < bv) { bv = v; bsel = l; }
    }
    head[bsel]++;
    int idx = (int)(u32)bv;
    outd[k] = unfkey((u32)(bv >> 32));
    outl[k] = (float)ref_y[idx];
  }
}

extern "C" void kernel_launch(void* const* d_in, const int* in_sizes, int n_in,
                              void* d_out, int out_size, void* d_ws, size_t ws_size,
                              hipStream_t stream) {
  const float* x     = (const float*)d_in[0];   // 2048*25
  const float* ref_x = (const float*)d_in[1];   // 200000*8
  const int*   ref_y = (const int*)d_in[2];     // 200000
  const float* Wc    = (const float*)d_in[3];   // 16*25*8
  const float* bc    = (const float*)d_in[4];   // 16
  const float* Wl    = (const float*)d_in[5];   // 8*16
  const float* bl    = (const float*)d_in[6];   // 8

  float* q   = (float*)d_ws;                    // 2048*8
  float* qq  = q + BATCH * 8;                   // 2048
  u64*   ck  = (u64*)(qq + BATCH + BATCH);      // keys: 2048*128 u64 (8B-aligned pad)

  qk<<<BATCH / 256, 256, 0, stream>>>(x, Wc, bc, Wl, bl, q, qq);
  distk<<<dim3(NSPLIT, BATCH / 16), 256, 0, stream>>>(q, qq, ref_x, ck);
  mergek<<<BATCH / 256, 256, 0, stream>>>(ck, ref_y, (float*)d_out);
}